// MaxPool2d_51719996178968
// MI455X (gfx1250) — compile-verified
//
#include <hip/hip_runtime.h>
#include <stdint.h>

typedef float v4f __attribute__((ext_vector_type(4)));

#define IN_W   224
#define GRPS_PER_ROW 28   // 4 outputs (8 input cols) per thread, 112/4 = 28

// Each thread: async-DMA four 16B chunks (2 rows x 32B) into its private 64B
// LDS slot, wait on ASYNCcnt, reduce 16 floats -> 4 outputs, NT-store 16B.
//
// Address identities (no per-plane math needed):
//   output row r = plane*112 + oy  ->  input offset = 448*r + col
//   idx = r*28 + grp               ->  in  = 16*idx - 8*grp ; out = 4*idx
__global__ __launch_bounds__(256) void maxpool2x2_kernel(
    const float* __restrict__ x, float* __restrict__ y, unsigned total_grps) {

    // 4 x float4 per thread, 16B aligned
    __shared__ v4f sh4[256 * 4];

    const unsigned tid = threadIdx.x;
    const unsigned idx = blockIdx.x * 256u + tid;

    // Workgroup-relative LDS byte offset of this lane's 64B slot.
    const unsigned lds0 = (unsigned)(uintptr_t)(&sh4[tid * 4]);
    const unsigned lds2 = lds0 + 32u;

    if (idx < total_grps) {
        const unsigned q   = idx / GRPS_PER_ROW;
        const unsigned grp = idx - q * GRPS_PER_ROW;

        // Row 2r chunk start: x + 448*r + 8*grp == x + 16*idx - 8*grp
        const float* g0 = x + ((size_t)idx * 16u) - (size_t)(grp * 8u);
        const float* g1 = g0 + IN_W;   // row 2r+1, +896 bytes

        // CDNA5 async global->LDS DMA loads (ASYNCcnt). The immediate offset
        // is added to BOTH the LDS and the global address, so each row needs
        // only one LDS-base VGPR and one global pointer.
        asm volatile("global_load_async_to_lds_b128 %0, %1, off"
                     :: "v"(lds0), "v"(g0) : "memory");
        asm volatile("global_load_async_to_lds_b128 %0, %1, off offset:16"
                     :: "v"(lds0), "v"(g0) : "memory");
        asm volatile("global_load_async_to_lds_b128 %0, %1, off"
                     :: "v"(lds2), "v"(g1) : "memory");
        asm volatile("global_load_async_to_lds_b128 %0, %1, off offset:16"
                     :: "v"(lds2), "v"(g1) : "memory");
    }

    // Wait for this wave's async DMA writes to LDS to complete.
    asm volatile("s_wait_asynccnt 0" ::: "memory");

    if (idx < total_grps) {
        const v4f a0 = sh4[tid * 4 + 0];   // row 2r,   cols 0..3  of the chunk
        const v4f a1 = sh4[tid * 4 + 1];   // row 2r,   cols 4..7
        const v4f b0 = sh4[tid * 4 + 2];   // row 2r+1, cols 0..3
        const v4f b1 = sh4[tid * 4 + 3];   // row 2r+1, cols 4..7

        v4f r;
        r.x = fmaxf(fmaxf(a0.x, a0.y), fmaxf(b0.x, b0.y));
        r.y = fmaxf(fmaxf(a0.z, a0.w), fmaxf(b0.z, b0.w));
        r.z = fmaxf(fmaxf(a1.x, a1.y), fmaxf(b1.x, b1.y));
        r.w = fmaxf(fmaxf(a1.z, a1.w), fmaxf(b1.z, b1.w));

        // Streaming output: non-temporal 128-bit store, 16B-aligned.
        v4f* dst = (v4f*)(y + (size_t)idx * 4u);
        __builtin_nontemporal_store(r, dst);
    }
}

extern "C" void kernel_launch(void* const* d_in, const int* in_sizes, int n_in,
                              void* d_out, int out_size, void* d_ws, size_t ws_size,
                              hipStream_t stream) {
    (void)in_sizes; (void)n_in; (void)d_ws; (void)ws_size;

    const float* x = (const float*)d_in[0];
    float*       y = (float*)d_out;

    // out_size = 32*64*112*112 = 25,690,112 ; four outputs per thread
    const unsigned total_grps = (unsigned)(out_size / 4);
    const unsigned block = 256;
    const unsigned grid  = (total_grps + block - 1) / block;  // 25,088

    maxpool2x2_kernel<<<grid, block, 0, stream>>>(x, y, total_grps);
}